// Hake_65369402245832
// MI455X (gfx1250) — compile-verified
//
#include <hip/hip_runtime.h>
#include <math.h>

typedef float v2f __attribute__((ext_vector_type(2)));
typedef float v8f __attribute__((ext_vector_type(8)));

#define N_NODES 100000
#define N_RELS  500
#define N_EDGES 1000000
#define DDIM    64

// ---------------------------------------------------------------------------
// Kernel 1: relation tables (500 x 64). Tiny: 500*64 threads, two 64-dots each.
//   relabs[r,d] = |g[r,d]|
//   bias[r,d]   = min(relabs[r,:] @ W_bias[d,:] + b_bias[d], 1), masked -> relabs
//   Arel[r,d]   = acos(relabs[r,:] @ W_phase[d,:] + b_phase[d])
// ---------------------------------------------------------------------------
__global__ void hake_rel_kernel(const float* __restrict__ g,
                                const float* __restrict__ Wb,
                                const float* __restrict__ bb,
                                const float* __restrict__ Wp,
                                const float* __restrict__ bp,
                                float* __restrict__ relabs,
                                float* __restrict__ biasT,
                                float* __restrict__ Arel) {
    int t = blockIdx.x * blockDim.x + threadIdx.x;
    if (t >= N_RELS * DDIM) return;
    int r = t >> 6;
    int d = t & 63;
    const float* gr = g + r * DDIM;
    const float* wbr = Wb + d * DDIM;
    const float* wpr = Wp + d * DDIM;
    float accB = 0.f, accP = 0.f;
#pragma unroll 8
    for (int k = 0; k < DDIM; ++k) {
        float ra = fabsf(gr[k]);
        accB = fmaf(ra, wbr[k], accB);
        accP = fmaf(ra, wpr[k], accP);
    }
    float ra_d = fabsf(gr[d]);
    float bv = fminf(accB + bb[d], 1.0f);
    if (bv < -ra_d) bv = ra_d;
    float pv = fminf(fmaxf(accP + bp[d], -1.0f), 1.0f);
    relabs[t] = ra_d;
    biasT[t]  = bv;
    Arel[t]   = acosf(pv);
}

// ---------------------------------------------------------------------------
// Kernel 2: node phase table via WMMA f32 16x16x4.
//   Anode[n,d] = acos( (h @ W_phase^T)[n,d] + b_phase[d] )
// One wave per 16x16 output tile; K=64 in 16 steps of 4.
// A layout (16x4, MxK): lane&15 = M; VGPR pair holds K = 2*(lane>>4) + {0,1}.
// B layout (4x16, KxN): lane&15 = N; VGPR pair holds K = 2*(lane>>4) + {0,1}.
// C/D layout: element v of v8f holds row (v + 8*(lane>>4)), col (lane&15).
// ---------------------------------------------------------------------------
__global__ void __launch_bounds__(32)
hake_nodephase_kernel(const float* __restrict__ h,
                      const float* __restrict__ Wp,
                      const float* __restrict__ bp,
                      float* __restrict__ Anode) {
    const int tile  = blockIdx.x;          // 6250*4 tiles
    const int tileM = tile >> 2;           // node tile
    const int tileN = tile & 3;            // output-dim tile
    const int lane  = threadIdx.x;
    const int half  = lane >> 4;           // 0 or 1
    const int l15   = lane & 15;

    const float* arow = h  + (tileM * 16 + l15) * DDIM + 2 * half; // A: M=l15
    const float* brow = Wp + (tileN * 16 + l15) * DDIM + 2 * half; // B: N=l15

    v8f c = {};
#pragma unroll
    for (int k0 = 0; k0 < DDIM; k0 += 4) {
        v2f a = *(const v2f*)(arow + k0);
        v2f b = *(const v2f*)(brow + k0);
        c = __builtin_amdgcn_wmma_f32_16x16x4_f32(
            /*neg_a=*/false, a, /*neg_b=*/false, b,
            /*c_mod=*/(short)0, c, /*reuse_a=*/false, /*reuse_b=*/false);
    }

    const int colg = tileN * 16 + l15;
    const float bpv = bp[colg];
#pragma unroll
    for (int v = 0; v < 8; ++v) {
        int rowg = tileM * 16 + v + 8 * half;
        float val = fminf(fmaxf(c[v] + bpv, -1.0f), 1.0f);
        Anode[rowg * DDIM + colg] = acosf(val);
    }
}

// ---------------------------------------------------------------------------
// Kernel 3: per-edge scoring. One wave32 per edge, 2 dims per lane.
// All gathered tables (h 25.6MB, Anode 25.6MB, rel tables 384KB) fit in L2.
// ---------------------------------------------------------------------------
__global__ void __launch_bounds__(256)
hake_edge_kernel(const float* __restrict__ h,
                 const int*   __restrict__ edge_idx,   // [2, E] flat
                 const int*   __restrict__ edge_type,
                 const float* __restrict__ relabs,
                 const float* __restrict__ biasT,
                 const float* __restrict__ Arel,
                 const float* __restrict__ Anode,
                 const float* __restrict__ lambda_p,
                 float* __restrict__ out) {
    const int wave = threadIdx.x >> 5;
    const int lane = threadIdx.x & 31;
    const int e = blockIdx.x * 8 + wave;
    if (e >= N_EDGES) return;

    const int row = edge_idx[e];
    const int col = edge_idx[N_EDGES + e];
    const int et  = edge_type[e];

    const float* hh = h      + row * DDIM;
    const float* ht = h      + col * DDIM;
    const float* Ah = Anode  + row * DDIM;
    const float* At = Anode  + col * DDIM;
    const float* ra = relabs + et  * DDIM;
    const float* bi = biasT  + et  * DDIM;
    const float* Ar = Arel   + et  * DDIM;

    float accM = 0.f, accP = 0.f;
#pragma unroll
    for (int j = 0; j < 2; ++j) {
        int d = lane + 32 * j;
        float hd = hh[d], td = ht[d];
        float r  = ra[d], b  = bi[d];
        float m  = hd * r + (hd + td) * b - td;
        accM = fmaf(m, m, accM);
        float ang = Ah[d] + Ar[d] - At[d];
        accP += fabsf(__sinf(ang));
    }
    // wave32 butterfly reduction
#pragma unroll
    for (int off = 16; off > 0; off >>= 1) {
        accM += __shfl_xor(accM, off, 32);
        accP += __shfl_xor(accP, off, 32);
    }
    if (lane == 0) {
        out[e] = sqrtf(accM) + lambda_p[0] * accP;
    }
}

// ---------------------------------------------------------------------------
extern "C" void kernel_launch(void* const* d_in, const int* in_sizes, int n_in,
                              void* d_out, int out_size, void* d_ws, size_t ws_size,
                              hipStream_t stream) {
    const float* h        = (const float*)d_in[0];
    const float* g        = (const float*)d_in[1];
    const int*   edge_idx = (const int*)  d_in[2];
    const int*   edge_ty  = (const int*)  d_in[3];
    const float* W_bias   = (const float*)d_in[4];
    const float* b_bias   = (const float*)d_in[5];
    const float* W_phase  = (const float*)d_in[6];
    const float* b_phase  = (const float*)d_in[7];
    const float* lambda_p = (const float*)d_in[8];
    float* out = (float*)d_out;

    // Workspace layout (floats): Anode [N_NODES*64] | relabs | biasT | Arel [500*64 each]
    float* Anode  = (float*)d_ws;
    float* relabs = Anode  + (size_t)N_NODES * DDIM;
    float* biasT  = relabs + (size_t)N_RELS * DDIM;
    float* Arel   = biasT  + (size_t)N_RELS * DDIM;

    // 1) relation tables: 500*64 = 32000 threads
    hake_rel_kernel<<<(N_RELS * DDIM + 255) / 256, 256, 0, stream>>>(
        g, W_bias, b_bias, W_phase, b_phase, relabs, biasT, Arel);

    // 2) node phase GEMM+acos: (100000/16) * (64/16) = 25000 tiles, 1 wave each
    hake_nodephase_kernel<<<(N_NODES / 16) * (DDIM / 16), 32, 0, stream>>>(
        h, W_phase, b_phase, Anode);

    // 3) edge scoring: 8 edges per 256-thread block
    hake_edge_kernel<<<N_EDGES / 8, 256, 0, stream>>>(
        h, edge_idx, edge_ty, relabs, biasT, Arel, Anode, lambda_p, out);
}